// SinkhornMatch_29326036697420
// MI455X (gfx1250) — compile-verified
//
#include <hip/hip_runtime.h>
#include <hip/hip_bf16.h>
#include <math.h>

typedef float v2f __attribute__((ext_vector_type(2)));
typedef float v8f __attribute__((ext_vector_type(8)));

#define DIMK 512
#define MM   4096
#define NN   4096
#define LD   4097      // (NN+1) augmented matrix leading dim
#define SITERS 100

#define KCHUNK 32            // k-elements staged per double-buffer step
#define BROWPAD 36           // LDS row stride (floats): 16B aligned, bank-conflict-free
#define BBUFBYTES (64 * BROWPAD * 4)

// ------------------------------------------------------------------
// GEMM:  C[i][j] = scale * ( sum_k A[i][k]*B[j][k] + bias[j] )
// Block: 8 waves -> 128(M) x 64(N) tile. B tile is shared by all waves:
// staged through LDS with GLOBAL_LOAD_ASYNC_TO_LDS_B128 (double buffered,
// s_wait_asynccnt). A fragments are per-wave: software-pipelined in regs.
// Per wave per k-step: 4x V_WMMA_F32_16X16X4_F32 (16x64 output strip).
// Requires K % 32 == 0, all lanes active (EXEC all ones for WMMA).
// ------------------------------------------------------------------
__global__ __launch_bounds__(256) void gemm_nt_wmma(
    const float* __restrict__ A, const float* __restrict__ B,
    const float* __restrict__ bias, float scale,
    float* __restrict__ C, int ldc,
    float* __restrict__ C2, int ldc2, int K)
{
  __shared__ float Bs[2][64 * BROWPAD];

  const int tid  = threadIdx.x;
  const int wave = tid >> 5;
  const int lane = tid & 31;
  const int lm   = lane & 15;          // M (A) / N (B) index within fragment
  const int hi   = lane >> 4;          // lane group
  const int kb   = hi * 2;             // K sub-offset within fragment
  const int row0 = blockIdx.y * 128 + wave * 16;
  const int col0 = blockIdx.x * 64;

  // ---- async B-tile loader setup: 64 rows x 32 floats per chunk,
  //      512 x 16B segments, 2 per thread ----
  const int brow0 = tid >> 3;          // 0..31
  const int brow1 = brow0 + 32;        // 32..63
  const int bcolf = (tid & 7) * 4;     // float offset of 16B segment in row
  const float* gb0 = B + (size_t)(col0 + brow0) * K + bcolf;
  const float* gb1 = B + (size_t)(col0 + brow1) * K + bcolf;
  const unsigned ldsBase = (unsigned)(size_t)&Bs[0][0];
  const unsigned lo0 = ldsBase + (unsigned)(brow0 * BROWPAD + bcolf) * 4u;
  const unsigned lo1 = ldsBase + (unsigned)(brow1 * BROWPAD + bcolf) * 4u;

  // prologue: chunk 0 -> buffer 0
  asm volatile("global_load_async_to_lds_b128 %0, %1, off"
               :: "v"(lo0), "v"(gb0) : "memory");
  asm volatile("global_load_async_to_lds_b128 %0, %1, off"
               :: "v"(lo1), "v"(gb1) : "memory");

  // ---- per-wave A fragment pipeline (8 v2f per 32-k chunk) ----
  const float* aptr = A + (size_t)(row0 + lm) * K + kb;
  v2f acur[8], anxt[8];
  #pragma unroll
  for (int q = 0; q < 8; ++q)
    acur[q] = *(const v2f*)(aptr + q * 4);

  v8f acc[4] = {};
  const int CHUNKS = K / KCHUNK;

  for (int c = 0; c < CHUNKS; ++c) {
    const bool more = (c + 1 < CHUNKS);
    if (more) {
      // issue async B loads for chunk c+1 into buffer (c+1)&1
      const unsigned boff = (unsigned)(((c + 1) & 1) * BBUFBYTES);
      const float* g0 = gb0 + (c + 1) * KCHUNK;
      const float* g1 = gb1 + (c + 1) * KCHUNK;
      asm volatile("global_load_async_to_lds_b128 %0, %1, off"
                   :: "v"(lo0 + boff), "v"(g0) : "memory");
      asm volatile("global_load_async_to_lds_b128 %0, %1, off"
                   :: "v"(lo1 + boff), "v"(g1) : "memory");
      // prefetch next A chunk into registers (overlaps with compute below)
      #pragma unroll
      for (int q = 0; q < 8; ++q)
        anxt[q] = *(const v2f*)(aptr + (c + 1) * KCHUNK + q * 4);
      // wait for chunk c only (the 2 newer async loads stay in flight)
      asm volatile("s_wait_asynccnt 0x2" ::: "memory");
    } else {
      asm volatile("s_wait_asynccnt 0x0" ::: "memory");
    }
    __syncthreads();   // all waves' portions of chunk c are in LDS

    const float* bb = &Bs[c & 1][0];
    #pragma unroll
    for (int q = 0; q < 8; ++q) {
      const int kk = q * 4;
      v2f a  = acur[q];
      v2f b0 = *(const v2f*)(bb + ( 0 + lm) * BROWPAD + kk + kb);
      v2f b1 = *(const v2f*)(bb + (16 + lm) * BROWPAD + kk + kb);
      v2f b2 = *(const v2f*)(bb + (32 + lm) * BROWPAD + kk + kb);
      v2f b3 = *(const v2f*)(bb + (48 + lm) * BROWPAD + kk + kb);
      acc[0] = __builtin_amdgcn_wmma_f32_16x16x4_f32(false, a, false, b0, (short)0, acc[0], false, false);
      acc[1] = __builtin_amdgcn_wmma_f32_16x16x4_f32(false, a, false, b1, (short)0, acc[1], false, false);
      acc[2] = __builtin_amdgcn_wmma_f32_16x16x4_f32(false, a, false, b2, (short)0, acc[2], false, false);
      acc[3] = __builtin_amdgcn_wmma_f32_16x16x4_f32(false, a, false, b3, (short)0, acc[3], false, false);
    }
    __syncthreads();   // done reading buffer (c&1) before it is refilled

    #pragma unroll
    for (int q = 0; q < 8; ++q) acur[q] = anxt[q];
  }

  // C/D layout: VGPR r, lanes 0-15 -> M=r, lanes 16-31 -> M=8+r; N = lane&15
  #pragma unroll
  for (int t = 0; t < 4; ++t) {
    const int col = col0 + t * 16 + lm;
    const float bv = bias ? bias[col] : 0.0f;
    #pragma unroll
    for (int r = 0; r < 8; ++r) {
      const int row = row0 + hi * 8 + r;
      const float val = (acc[t][r] + bv) * scale;
      C[(size_t)row * ldc + col] = val;
      if (C2) C2[(size_t)row * ldc2 + col] = val;
    }
  }
}

// ------------------------------------------------------------------
// Z0 dustbin fill: last column (incl. corner) and last row = alpha
// ------------------------------------------------------------------
__global__ void fill_dustbin(float* __restrict__ Z0, const float* __restrict__ alpha_p)
{
  const float a = *alpha_p;
  int idx = blockIdx.x * blockDim.x + threadIdx.x;
  if (idx <= MM) {                         // last column rows 0..MM (incl corner)
    Z0[(size_t)idx * LD + NN] = a;
  } else if (idx <= MM + NN) {             // last row cols 0..NN-1
    int j = idx - (MM + 1);
    Z0[(size_t)MM * LD + j] = a;
  }
}

__global__ void zero_f32(float* __restrict__ p, size_t n)
{
  size_t idx = (size_t)blockIdx.x * blockDim.x + threadIdx.x;
  if (idx < n) p[idx] = 0.0f;
}

// ------------------------------------------------------------------
// Online logsumexp helpers
// ------------------------------------------------------------------
__device__ __forceinline__ void lse_acc(float x, float& m, float& s)
{
  if (x > m) { s = s * __expf(m - x) + 1.0f; m = x; }
  else       { s += __expf(x - m); }
}
__device__ __forceinline__ void lse_join(float m2, float s2, float& m, float& s)
{
  float mm = fmaxf(m, m2);
  if (mm == -INFINITY) return;
  s = s * __expf(m - mm) + s2 * __expf(m2 - mm);
  m = mm;
}

// t[i] = target(i) - logsumexp_j( Z[i][j] + w[j] ),  j in [0, LD)
__global__ __launch_bounds__(256) void lse_row(
    const float* __restrict__ Z, const float* __restrict__ w,
    float* __restrict__ t, float tmain, float tlast)
{
  __shared__ float sm[256], ss[256];
  const int i = blockIdx.x;
  const float* zr = Z + (size_t)i * LD;
  float m = -INFINITY, s = 0.0f;
  for (int j = threadIdx.x; j < LD; j += 256)
    lse_acc(zr[j] + w[j], m, s);
  sm[threadIdx.x] = m; ss[threadIdx.x] = s;
  __syncthreads();
  for (int off = 128; off > 0; off >>= 1) {
    if (threadIdx.x < off) {
      lse_join(sm[threadIdx.x + off], ss[threadIdx.x + off], m, s);
      sm[threadIdx.x] = m; ss[threadIdx.x] = s;
    }
    __syncthreads();
  }
  if (threadIdx.x == 0)
    t[i] = ((i == LD - 1) ? tlast : tmain) - (sm[0] + __logf(ss[0]));
}

// Column-pass partials: thread = one column, rows chunked for parallelism.
#define COLCHUNKS 32
#define ROWS_PER_CHUNK 129   // 32*129 = 4128 >= 4097
__global__ __launch_bounds__(256) void lse_col_partial(
    const float* __restrict__ Z, const float* __restrict__ u,
    float* __restrict__ pm, float* __restrict__ ps)
{
  const int j = blockIdx.x * 256 + threadIdx.x;
  if (j >= LD) return;
  const int r0 = blockIdx.y * ROWS_PER_CHUNK;
  const int r1 = min(r0 + ROWS_PER_CHUNK, LD);
  float m = -INFINITY, s = 0.0f;
  for (int i = r0; i < r1; ++i)
    lse_acc(Z[(size_t)i * LD + j] + u[i], m, s);
  pm[(size_t)blockIdx.y * LD + j] = m;
  ps[(size_t)blockIdx.y * LD + j] = s;
}

__global__ __launch_bounds__(256) void lse_col_combine(
    const float* __restrict__ pm, const float* __restrict__ ps,
    float* __restrict__ v, float tmain, float tlast)
{
  const int j = blockIdx.x * 256 + threadIdx.x;
  if (j >= LD) return;
  float m = -INFINITY, s = 0.0f;
  for (int r = 0; r < COLCHUNKS; ++r)
    lse_join(pm[(size_t)r * LD + j], ps[(size_t)r * LD + j], m, s);
  v[j] = ((j == LD - 1) ? tlast : tmain) - (m + __logf(s));
}

// ------------------------------------------------------------------
// scores = exp(Z0 + u + v - norm)  over full (LD x LD)
// ------------------------------------------------------------------
__global__ __launch_bounds__(256) void scores_kernel(
    const float* __restrict__ Z0, const float* __restrict__ u,
    const float* __restrict__ v, float norm, float* __restrict__ out)
{
  const int j = blockIdx.x * 256 + threadIdx.x;
  if (j >= LD) return;
  const int i = blockIdx.y;
  const size_t idx = (size_t)i * LD + j;
  out[idx] = __expf(Z0[idx] + u[i] + v[j] - norm);
}

// ------------------------------------------------------------------
// Top-3 helpers (sorted descending triple insert)
// ------------------------------------------------------------------
__device__ __forceinline__ void ins3(float x, int j,
                                     float& t0, float& t1, float& t2,
                                     int& i0, int& i1, int& i2)
{
  if (x > t0)      { t2 = t1; i2 = i1; t1 = t0; i1 = i0; t0 = x; i0 = j; }
  else if (x > t1) { t2 = t1; i2 = i1; t1 = x;  i1 = j; }
  else if (x > t2) { t2 = x;  i2 = j; }
}

// Row-wise top-3 over logits (monotonic with scores): value = Z0[i][j] + v[j]
__global__ __launch_bounds__(256) void topk_row_kernel(
    const float* __restrict__ Z0, const float* __restrict__ v,
    int* __restrict__ rowtop)
{
  __shared__ float sv[256 * 3];
  __shared__ int   si[256 * 3];
  const int i = blockIdx.x;
  const float* zr = Z0 + (size_t)i * LD;
  float t0 = -INFINITY, t1 = -INFINITY, t2 = -INFINITY;
  int i0 = 0, i1 = 0, i2 = 0;
  for (int j = threadIdx.x; j < NN; j += 256)
    ins3(zr[j] + v[j], j, t0, t1, t2, i0, i1, i2);
  sv[threadIdx.x * 3 + 0] = t0; si[threadIdx.x * 3 + 0] = i0;
  sv[threadIdx.x * 3 + 1] = t1; si[threadIdx.x * 3 + 1] = i1;
  sv[threadIdx.x * 3 + 2] = t2; si[threadIdx.x * 3 + 2] = i2;
  __syncthreads();
  for (int off = 128; off > 0; off >>= 1) {
    if (threadIdx.x < off) {
      const int o = (threadIdx.x + off) * 3;
      ins3(sv[o + 0], si[o + 0], t0, t1, t2, i0, i1, i2);
      ins3(sv[o + 1], si[o + 1], t0, t1, t2, i0, i1, i2);
      ins3(sv[o + 2], si[o + 2], t0, t1, t2, i0, i1, i2);
      sv[threadIdx.x * 3 + 0] = t0; si[threadIdx.x * 3 + 0] = i0;
      sv[threadIdx.x * 3 + 1] = t1; si[threadIdx.x * 3 + 1] = i1;
      sv[threadIdx.x * 3 + 2] = t2; si[threadIdx.x * 3 + 2] = i2;
    }
    __syncthreads();
  }
  if (threadIdx.x == 0) {
    rowtop[i * 3 + 0] = i0; rowtop[i * 3 + 1] = i1; rowtop[i * 3 + 2] = i2;
  }
}

// Column-wise top-3: value = Z0[i][j] + u[i]; chunked over rows, coalesced.
#define TCHUNKS 16
#define TROWS   256   // 16*256 = 4096 rows
__global__ __launch_bounds__(256) void topk_col_partial(
    const float* __restrict__ Z0, const float* __restrict__ u,
    float* __restrict__ pv, int* __restrict__ pi)
{
  const int j = blockIdx.x * 256 + threadIdx.x;
  if (j >= NN) return;
  const int r0 = blockIdx.y * TROWS;
  float t0 = -INFINITY, t1 = -INFINITY, t2 = -INFINITY;
  int i0 = 0, i1 = 0, i2 = 0;
  for (int i = r0; i < r0 + TROWS; ++i)
    ins3(Z0[(size_t)i * LD + j] + u[i], i, t0, t1, t2, i0, i1, i2);
  const size_t base = ((size_t)blockIdx.y * NN + j) * 3;
  pv[base + 0] = t0; pi[base + 0] = i0;
  pv[base + 1] = t1; pi[base + 1] = i1;
  pv[base + 2] = t2; pi[base + 2] = i2;
}

__global__ __launch_bounds__(256) void topk_col_combine(
    const float* __restrict__ pv, const int* __restrict__ pi,
    int* __restrict__ coltop)
{
  const int j = blockIdx.x * 256 + threadIdx.x;
  if (j >= NN) return;
  float t0 = -INFINITY, t1 = -INFINITY, t2 = -INFINITY;
  int i0 = 0, i1 = 0, i2 = 0;
  for (int r = 0; r < TCHUNKS; ++r) {
    const size_t base = ((size_t)r * NN + j) * 3;
    ins3(pv[base + 0], pi[base + 0], t0, t1, t2, i0, i1, i2);
    ins3(pv[base + 1], pi[base + 1], t0, t1, t2, i0, i1, i2);
    ins3(pv[base + 2], pi[base + 2], t0, t1, t2, i0, i1, i2);
  }
  coltop[j * 3 + 0] = i0; coltop[j * 3 + 1] = i1; coltop[j * 3 + 2] = i2;
}

// assignment[i][j] = 1 iff j in rowtop(i), i in coltop(j), score > threshold
__global__ __launch_bounds__(256) void mutual_kernel(
    const float* __restrict__ Z0, const float* __restrict__ u,
    const float* __restrict__ v, const int* __restrict__ rowtop,
    const int* __restrict__ coltop, float logthr, float norm,
    float* __restrict__ out)
{
  const int idx = blockIdx.x * blockDim.x + threadIdx.x;
  if (idx >= MM * 3) return;
  const int i = idx / 3;
  const int j = rowtop[idx];
  const float logit = Z0[(size_t)i * LD + j] + u[i] + v[j] - norm;
  if (logit > logthr &&
      (coltop[j * 3 + 0] == i || coltop[j * 3 + 1] == i || coltop[j * 3 + 2] == i))
    out[(size_t)i * NN + j] = 1.0f;
}

// ------------------------------------------------------------------
// Host launcher
// ------------------------------------------------------------------
extern "C" void kernel_launch(void* const* d_in, const int* in_sizes, int n_in,
                              void* d_out, int out_size, void* d_ws, size_t ws_size,
                              hipStream_t stream)
{
  const float* desc0 = (const float*)d_in[0];   // (4096, 512)
  const float* desc1 = (const float*)d_in[1];   // (4096, 512)
  const float* W     = (const float*)d_in[2];   // (512, 512)
  const float* bias  = (const float*)d_in[3];   // (512,)
  const float* alpha = (const float*)d_in[4];   // scalar

  float* out_kn     = (float*)d_out;                         // 4096*4096
  float* out_scores = out_kn + (size_t)MM * NN;              // 4097*4097
  float* out_assign = out_scores + (size_t)LD * LD;          // 4096*4096

  // workspace layout (floats) — ~87 MB total
  float* wsf    = (float*)d_ws;
  float* m0s    = wsf;                                   // 4096*512
  float* m1s    = m0s + (size_t)MM * DIMK;               // 4096*512
  float* Z0     = m1s + (size_t)NN * DIMK;               // 4097*4097
  float* uvec   = Z0 + (size_t)LD * LD;                  // 4097
  float* vvec   = uvec + LD;                             // 4097
  float* pm     = vvec + LD;                             // 32*4097
  float* ps     = pm + (size_t)COLCHUNKS * LD;           // 32*4097
  int*   rowtop = (int*)(ps + (size_t)COLCHUNKS * LD);   // 4096*3
  float* pcv    = (float*)(rowtop + (size_t)MM * 3);     // 16*4096*3
  int*   pci    = (int*)(pcv + (size_t)TCHUNKS * NN * 3);// 16*4096*3
  int*   coltop = pci + (size_t)TCHUNKS * NN * 3;        // 4096*3

  const float INVS   = 1.0f / powf((float)DIMK, 0.25f);  // 1/s, s = d^0.25
  const float NORM   = -logf((float)(MM + NN));
  const float TLAST  = logf((float)NN) + NORM;           // log(n)+norm (m==n)
  const float LOGTHR = logf(0.05f);

  // 1) projections: m0s = (desc @ W^T + b)/s,  m1s likewise
  gemm_nt_wmma<<<dim3(DIMK / 64, MM / 128), 256, 0, stream>>>(
      desc0, W, bias, INVS, m0s, DIMK, nullptr, 0, DIMK);
  gemm_nt_wmma<<<dim3(DIMK / 64, NN / 128), 256, 0, stream>>>(
      desc1, W, bias, INVS, m1s, DIMK, nullptr, 0, DIMK);

  // 2) Kn = m0s @ m1s^T  -> out_kn and Z0 interior
  gemm_nt_wmma<<<dim3(NN / 64, MM / 128), 256, 0, stream>>>(
      m0s, m1s, nullptr, 1.0f, out_kn, NN, Z0, LD, DIMK);

  // 3) dustbin row/col = alpha; u=v=0
  fill_dustbin<<<(2 * MM + 1 + 255) / 256, 256, 0, stream>>>(Z0, alpha);
  zero_f32<<<(2 * LD + 255) / 256, 256, 0, stream>>>(uvec, (size_t)2 * LD);

  // 4) Sinkhorn iterations (Z0 is L2-resident: 67 MB < 192 MB)
  for (int it = 0; it < SITERS; ++it) {
    lse_row<<<LD, 256, 0, stream>>>(Z0, vvec, uvec, NORM, TLAST);
    lse_col_partial<<<dim3((LD + 255) / 256, COLCHUNKS), 256, 0, stream>>>(Z0, uvec, pm, ps);
    lse_col_combine<<<(LD + 255) / 256, 256, 0, stream>>>(pm, ps, vvec, NORM, TLAST);
  }

  // 5) scores = exp(Z0 + u + v - norm)
  scores_kernel<<<dim3((LD + 255) / 256, LD), 256, 0, stream>>>(Z0, uvec, vvec, NORM, out_scores);

  // 6) mutual top-3 with threshold on the 4096x4096 interior
  topk_row_kernel<<<MM, 256, 0, stream>>>(Z0, vvec, rowtop);
  topk_col_partial<<<dim3(NN / 256, TCHUNKS), 256, 0, stream>>>(Z0, uvec, pcv, pci);
  topk_col_combine<<<NN / 256, 256, 0, stream>>>(pcv, pci, coltop);
  zero_f32<<<(int)(((size_t)MM * NN + 255) / 256), 256, 0, stream>>>(out_assign, (size_t)MM * NN);
  mutual_kernel<<<(MM * 3 + 255) / 256, 256, 0, stream>>>(
      Z0, uvec, vvec, rowtop, coltop, LOGTHR, NORM, out_assign);
}